// GSMLinear_55387898249344
// MI455X (gfx1250) — compile-verified
//
#include <hip/hip_runtime.h>
#include <stdint.h>

#define B_   256
#define OUT_ 512
#define IN_  512
#define NTOT ((uint32_t)B_ * OUT_ * IN_)       // 2^26
#define HALF (NTOT >> 1)                       // 2^25  -> b offset 128
#define N1   ((uint32_t)OUT_ * IN_)            // 2^18
#define H1   (N1 >> 1)

typedef float v2f __attribute__((ext_vector_type(2)));
typedef float v8f __attribute__((ext_vector_type(8)));

struct U2 { uint32_t x, y; };

__device__ __forceinline__ uint32_t rotl32(uint32_t v, int r) {
    return (v << r) | (v >> (32 - r));
}

// Exact threefry2x32 (JAX default PRNG core).
__device__ __forceinline__ U2 threefry2x32(uint32_t k0, uint32_t k1,
                                           uint32_t c0, uint32_t c1) {
    uint32_t ks2 = k0 ^ k1 ^ 0x1BD11BDAu;
    uint32_t x0 = c0 + k0, x1 = c1 + k1;
#define TFR(r) { x0 += x1; x1 = rotl32(x1, r); x1 ^= x0; }
    TFR(13) TFR(15) TFR(26) TFR(6)   x0 += k1;  x1 += ks2 + 1u;
    TFR(17) TFR(29) TFR(16) TFR(24)  x0 += ks2; x1 += k0  + 2u;
    TFR(13) TFR(15) TFR(26) TFR(6)   x0 += k0;  x1 += k1  + 3u;
    TFR(17) TFR(29) TFR(16) TFR(24)  x0 += k1;  x1 += ks2 + 4u;
    TFR(13) TFR(15) TFR(26) TFR(6)   x0 += ks2; x1 += k0  + 5u;
#undef TFR
    U2 r; r.x = x0; r.y = x1; return r;
}

// kg1/kg2 = jax.random.split(jax.random.key(42)); constant-foldable.
__device__ __forceinline__ void jax_split_keys(uint32_t& k1a, uint32_t& k1b,
                                               uint32_t& k2a, uint32_t& k2b) {
    U2 p = threefry2x32(0u, 42u, 0u, 2u);   // pairs (0,2) and (1,3)
    U2 q = threefry2x32(0u, 42u, 1u, 3u);
    k1a = p.x; k1b = q.x;                   // kg1
    k2a = p.y; k2b = q.y;                   // kg2
}

// JAX uniform(minval=tiny, maxval=1): bits>>9 | 1.0f exponent, -1, rescale.
__device__ __forceinline__ float to_unif(uint32_t bits) {
    uint32_t m = (bits >> 9) | 0x3f800000u;
    float f = __uint_as_float(m) - 1.0f;
    const float tiny = 1.1754944e-38f;
    return f * (1.0f - tiny) + tiny;
}

// exp(gumbel(u)) = 1 / (-ln u)
__device__ __forceinline__ float exp_gumbel(uint32_t bits) {
    float u = to_unif(bits);
    return __builtin_amdgcn_rcpf(-__logf(u));
}

__device__ __forceinline__ float sel8(const v8f& c, int j) {
    switch (j & 7) {
        case 0: return c[0]; case 1: return c[1];
        case 2: return c[2]; case 3: return c[3];
        case 4: return c[4]; case 5: return c[5];
        case 6: return c[6]; default: return c[7];
    }
}

// ---------- Kernel A: per-row argmax of logits+g1, coef[o], EL[o,i] ----------
__global__ void __launch_bounds__(256)
gsm_prep_kernel(const float* __restrict__ logits, const float* __restrict__ weight,
                float* __restrict__ EL, float* __restrict__ coef,
                int* __restrict__ istar) {
    uint32_t k1a, k1b, k2a, k2b;
    jax_split_keys(k1a, k1b, k2a, k2b);

    const int wave = (blockIdx.x * blockDim.x + threadIdx.x) >> 5; // 0..511
    const int lane = threadIdx.x & 31;
    const int o = wave;
    const float* lrow = logits + (size_t)o * IN_;

    float zmax = -3.4e38f;
    int   imax = 0;
    for (int t = 0; t < IN_ / 32; ++t) {
        const int i = lane + t * 32;
        const float lg = lrow[i];
        EL[(size_t)o * IN_ + i] = __expf(lg);

        const uint32_t idx = (uint32_t)o * IN_ + (uint32_t)i;   // < 2^18
        U2 tf = (idx < H1) ? threefry2x32(k1a, k1b, idx, idx + H1)
                           : threefry2x32(k1a, k1b, idx - H1, idx);
        const uint32_t bits = (idx < H1) ? tf.x : tf.y;
        const float u = to_unif(bits);
        const float g = -__logf(-__logf(u));
        const float z = lg + g;
        if (z > zmax) { zmax = z; imax = i; }    // first max kept (argmax tie rule)
    }
    // wave32 argmax reduction (lowest index on ties)
    for (int off = 16; off > 0; off >>= 1) {
        float oz = __shfl_down(zmax, off, 32);
        int   oi = __shfl_down(imax, off, 32);
        if (oz > zmax || (oz == zmax && oi < imax)) { zmax = oz; imax = oi; }
    }
    if (lane == 0) {
        const float w = weight[(size_t)o * IN_ + imax];
        const float weff = 2.0f * tanhf(2.0f * tanhf(w));
        coef[o]  = weff * __expf(lrow[imax]);     // includes exp(logits[o,i*])
        istar[o] = imax;
    }
}

// ---------- Kernel B: Den[b,o] via WMMA ones-reduction + final gather ----------
// One block per o (8 waves = 8 b-tiles of 16, each wave also owns b+128 twin).
__global__ void __launch_bounds__(256)
gsm_main_kernel(const float* __restrict__ x, const float* __restrict__ EL,
                const float* __restrict__ coef, const int* __restrict__ istar,
                float* __restrict__ y) {
    uint32_t k1a, k1b, k2a, k2b;
    jax_split_keys(k1a, k1b, k2a, k2b);

    const int o     = blockIdx.x;          // 0..511
    const int wv    = threadIdx.x >> 5;    // 0..7  -> b tile
    const int lane  = threadIdx.x & 31;
    const int bl    = lane & 15;           // M row of the A matrix
    const int kh    = lane >> 4;           // K half: lanes 16-31 hold K=2,3

    // Stage the exp(logits) row once per block.
    __shared__ float sEL[IN_];
    for (int i = threadIdx.x; i < IN_; i += blockDim.x)
        sEL[i] = EL[(size_t)o * IN_ + i];
    __syncthreads();

    const int b_lo = wv * 16 + bl;                       // 0..127
    const uint32_t nbase = ((uint32_t)b_lo * OUT_ + (uint32_t)o) * IN_; // < HALF

    v8f c_lo = {0.f,0.f,0.f,0.f,0.f,0.f,0.f,0.f};
    v8f c_hi = {0.f,0.f,0.f,0.f,0.f,0.f,0.f,0.f};
    const v2f ones = {1.0f, 1.0f};

    for (int i0 = 0; i0 < IN_; i0 += 4) {
        const int ia = i0 + kh * 2;                      // this lane's K pair
        // threefry pairs (n, n+HALF) -> uniforms for b_lo and b_lo+128 at once
        U2 t0 = threefry2x32(k2a, k2b, nbase + ia,     nbase + ia + HALF);
        U2 t1 = threefry2x32(k2a, k2b, nbase + ia + 1, nbase + ia + 1 + HALF);
        const float el0 = sEL[ia], el1 = sEL[ia + 1];
        v2f a_lo = { el0 * exp_gumbel(t0.x), el1 * exp_gumbel(t1.x) };
        v2f a_hi = { el0 * exp_gumbel(t0.y), el1 * exp_gumbel(t1.y) };
        // C += A(16x4) x ones(4x16): every column of C accumulates Den[b]
        c_lo = __builtin_amdgcn_wmma_f32_16x16x4_f32(false, a_lo, false, ones,
                                                     (short)0, c_lo, false, false);
        c_hi = __builtin_amdgcn_wmma_f32_16x16x4_f32(false, a_hi, false, ones,
                                                     (short)0, c_hi, false, false);
    }

    // C layout: lanes 0-15 VGPR j holds M=j; lanes 16-31 VGPR j holds M=8+j.
    const bool writer = (lane < 8) || (lane >= 16 && lane < 24);
    if (writer) {
        const int j    = lane & 7;
        const int blw  = (lane < 8) ? lane : (lane - 8);     // b_local 0..15
        const int bwlo = wv * 16 + blw;                      // 0..127
        const int bwhi = bwlo + 128;
        const float den_lo = sel8(c_lo, j);
        const float den_hi = sel8(c_hi, j);

        const int   is = istar[o];
        const float cf = coef[o];
        const uint32_t ns = ((uint32_t)bwlo * OUT_ + (uint32_t)o) * IN_ + (uint32_t)is;
        U2 ts = threefry2x32(k2a, k2b, ns, ns + HALF);
        const float rlo = exp_gumbel(ts.x);
        const float rhi = exp_gumbel(ts.y);

        y[(size_t)bwlo * OUT_ + o] = cf * x[(size_t)bwlo * IN_ + is] * rlo / den_lo;
        y[(size_t)bwhi * OUT_ + o] = cf * x[(size_t)bwhi * IN_ + is] * rhi / den_hi;
    }
}

extern "C" void kernel_launch(void* const* d_in, const int* in_sizes, int n_in,
                              void* d_out, int out_size, void* d_ws, size_t ws_size,
                              hipStream_t stream) {
    const float* x      = (const float*)d_in[0];   // [B, IN]
    const float* weight = (const float*)d_in[1];   // [OUT, IN]
    const float* logits = (const float*)d_in[2];   // [OUT, IN]
    float* y = (float*)d_out;                      // [B, OUT]

    // workspace: EL[OUT*IN] f32 | coef[OUT] f32 | istar[OUT] i32  (~1.05 MB)
    float* EL   = (float*)d_ws;
    float* coef = EL + (size_t)OUT_ * IN_;
    int*   istar = (int*)(coef + OUT_);

    // 512 waves, one per output row o
    gsm_prep_kernel<<<64, 256, 0, stream>>>(logits, weight, EL, coef, istar);
    // one block per o: 8 waves x (16 b-rows + their +128 twins) = all 256 b
    gsm_main_kernel<<<OUT_, 256, 0, stream>>>(x, EL, coef, istar, y);
}